// GATv2Star_55301998903766
// MI455X (gfx1250) — compile-verified
//
#include <hip/hip_runtime.h>
#include <math.h>

// ---- problem constants (match reference) ----
#define Bsz 32
#define Nsz 8192
#define IND 128          // IN_DIM
#define OUTD 128         // H*C
#define Hh 4
#define Cc 32
#define NEG_SLOPE 0.2f

// ---- WMMA vector types (CDNA5 f32 16x16x4) ----
typedef float v2f __attribute__((ext_vector_type(2)));
typedef float v8f __attribute__((ext_vector_type(8)));

// ---- workspace layout (floats) ----
// xl0    : [B][128]            = 4096
// stats  : [B*H][2] (max, Z)   = 256
// center : [B][128]            = 4096
// logits : [B*N][H]            = 1048576
#define WS_XL0    0
#define WS_STATS  4096
#define WS_CENTER (4096 + 256)
#define WS_LOGITS (4096 + 256 + 4096)

// =====================================================================
// K1: x_l0[b][col] = x[b,0,:] @ W_l[:,col] + b_l[col]   (32 x 128, tiny)
// =====================================================================
__global__ void gat_xl0_kernel(const float* __restrict__ x,
                               const float* __restrict__ Wl,
                               const float* __restrict__ bl,
                               float* __restrict__ ws) {
    const int b = blockIdx.x;       // 32 blocks
    const int col = threadIdx.x;    // 128 threads
    const float* xr = x + (size_t)b * Nsz * IND;   // row 0 of batch b
    float acc = bl[col];
    #pragma unroll 8
    for (int k = 0; k < IND; ++k)
        acc += xr[k] * Wl[k * OUTD + col];
    ws[WS_XL0 + b * OUTD + col] = acc;
}

// =====================================================================
// K2: main fused GEMM.  64 rows per block, 256 threads = 8 wave32.
//   - stage x tile (64x128) and W_r (128x128, K-paired) into LDS
//   - wave w: M-tile = w>>1, N-half = w&1 (4 N-tiles of 16 cols)
//   - 32 K-steps of v_wmma_f32_16x16x4_f32 per N-tile
//   - epilogue: +bias, store x_r to out, leakyrelu(xl0+xr)*att row-sum
//     via 16-lane butterflies -> logits[b,n,h]
// =====================================================================
__global__ void gat_gemm_kernel(const float* __restrict__ x,
                                const float* __restrict__ Wr,
                                const float* __restrict__ br,
                                const float* __restrict__ att,
                                float* __restrict__ out,
                                float* __restrict__ ws) {
    extern __shared__ float smem[];
    float* sx = smem;                 // 64*128 floats = 32 KB
    float* sw = smem + 64 * IND;      // 128*128 floats = 64 KB (K-paired)

    const int tid = threadIdx.x;
    const int rowbase = blockIdx.x * 64;
    const int b = blockIdx.x >> 7;    // 128 blocks per batch (8192/64)

    // ---- stage x tile: float4 copies ----
    {
        const float4* src = (const float4*)(x + (size_t)rowbase * IND);
        float4* dst = (float4*)sx;
        #pragma unroll
        for (int i = 0; i < (64 * IND / 4) / 256; ++i)
            dst[tid + i * 256] = src[tid + i * 256];
    }
    // ---- stage W_r in K-paired layout: sw2[kp*128 + n] = (W[2kp][n], W[2kp+1][n]) ----
    {
        float2* sw2 = (float2*)sw;
        for (int i = tid; i < (IND / 2) * OUTD; i += 256) {
            const int kp = i >> 7;
            const int n  = i & 127;
            float2 v;
            v.x = Wr[(2 * kp) * OUTD + n];
            v.y = Wr[(2 * kp + 1) * OUTD + n];
            sw2[kp * OUTD + n] = v;
        }
    }
    __syncthreads();

    const int lane  = tid & 31;
    const int wv    = tid >> 5;
    const int mtile = wv >> 1;        // 0..3
    const int nhalf = wv & 1;         // 0..1
    const int l15   = lane & 15;
    const int hi    = lane >> 4;      // 0 or 1
    const int koff  = hi * 2;         // K sub-offset per ISA f32 A/B layout

    const float* arow = sx + (mtile * 16 + l15) * IND;
    const float2* sw2 = (const float2*)sw;

    v8f acc[4];
    #pragma unroll
    for (int t = 0; t < 4; ++t)
        #pragma unroll
        for (int e = 0; e < 8; ++e) acc[t][e] = 0.0f;

    for (int kb = 0; kb < IND; kb += 4) {
        // A fragment: lane<16 -> K={kb,kb+1}, lane>=16 -> K={kb+2,kb+3}
        v2f a = *(const v2f*)(arow + kb + koff);
        #pragma unroll
        for (int t = 0; t < 4; ++t) {
            const int col = (nhalf * 4 + t) * 16 + l15;
            const float2 bw = sw2[((kb + koff) >> 1) * OUTD + col];
            v2f bf; bf.x = bw.x; bf.y = bw.y;
            acc[t] = __builtin_amdgcn_wmma_f32_16x16x4_f32(
                false, a, false, bf, (short)0, acc[t], false, false);
        }
    }

    // ---- fused epilogue ----
    const float* xl0 = ws + WS_XL0 + b * OUTD;
    float* logits = ws + WS_LOGITS;

    #pragma unroll
    for (int hh = 0; hh < 2; ++hh) {
        const int h = 2 * nhalf + hh;
        float rowsum[8];
        #pragma unroll
        for (int v = 0; v < 8; ++v) rowsum[v] = 0.0f;

        #pragma unroll
        for (int tt = 0; tt < 2; ++tt) {
            const int t   = hh * 2 + tt;
            const int col = (nhalf * 4 + t) * 16 + l15;
            const float bias = br[col];
            const float av   = att[col];   // att flat [H*C] == [128]
            const float xv   = xl0[col];
            #pragma unroll
            for (int v = 0; v < 8; ++v) {
                const float valb = acc[t][v] + bias;           // x_r (with bias)
                const int row = rowbase + mtile * 16 + v + hi * 8;
                out[(size_t)row * OUTD + col] = valb;
                const float z = xv + valb;                     // xl0 + xr
                const float e = (z > 0.0f) ? z : NEG_SLOPE * z;
                rowsum[v] += e * av;
            }
        }
        // butterfly reduce across the 16 lanes of each half-wave
        #pragma unroll
        for (int v = 0; v < 8; ++v) {
            float s = rowsum[v];
            s += __shfl_xor(s, 8, 16);
            s += __shfl_xor(s, 4, 16);
            s += __shfl_xor(s, 2, 16);
            s += __shfl_xor(s, 1, 16);
            if (l15 == 0) {
                const int row = rowbase + mtile * 16 + v + hi * 8;
                logits[(size_t)row * Hh + h] = s;
            }
        }
    }
}

// =====================================================================
// K3: per-(b,h) softmax stats (max, sum-exp) over N; zero center accum.
// =====================================================================
__global__ void gat_stats_kernel(float* __restrict__ ws) {
    __shared__ float red[256];
    const int bh = blockIdx.x;        // 128 blocks
    const int b = bh >> 2;
    const int h = bh & 3;
    const int tid = threadIdx.x;
    const float* lg = ws + WS_LOGITS + (size_t)b * Nsz * Hh + h;

    float mx = -INFINITY;
    for (int n = tid; n < Nsz; n += 256)
        mx = fmaxf(mx, lg[(size_t)n * Hh]);
    red[tid] = mx; __syncthreads();
    for (int s = 128; s > 0; s >>= 1) {
        if (tid < s) red[tid] = fmaxf(red[tid], red[tid + s]);
        __syncthreads();
    }
    mx = red[0]; __syncthreads();

    float se = 0.0f;
    for (int n = tid; n < Nsz; n += 256)
        se += expf(lg[(size_t)n * Hh] - mx);
    red[tid] = se; __syncthreads();
    for (int s = 128; s > 0; s >>= 1) {
        if (tid < s) red[tid] += red[tid + s];
        __syncthreads();
    }
    if (tid == 0) {
        ws[WS_STATS + bh * 2 + 0] = mx;
        ws[WS_STATS + bh * 2 + 1] = red[0];
    }
    // zero the center accumulator (one h-slice per batch does it)
    if (h == 0 && tid < OUTD) ws[WS_CENTER + b * OUTD + tid] = 0.0f;
}

// =====================================================================
// K4: center[b][col] += sum_n exp(logit[b,n,h]-max) * x_r[b,n,col]
//     grid = B*16 chunks of 512 rows; 256 threads: col = t&127, 2 row groups
// =====================================================================
__global__ void gat_agg_kernel(const float* __restrict__ out,
                               float* __restrict__ ws) {
    __shared__ float red[256];
    const int blk = blockIdx.x;           // 512 blocks
    const int b = blk >> 4;
    const int chunk = blk & 15;
    const int tid = threadIdx.x;
    const int col = tid & 127;
    const int rg = tid >> 7;              // 0/1
    const int h = col >> 5;
    const float mx = ws[WS_STATS + (b * Hh + h) * 2 + 0];
    const float* lg = ws + WS_LOGITS;

    float acc = 0.0f;
    const int rbase = chunk * 512 + rg;
    for (int i = 0; i < 256; ++i) {
        const int n = rbase + 2 * i;
        const size_t r = (size_t)b * Nsz + n;
        const float w = expf(lg[r * Hh + h] - mx);
        acc += w * out[r * OUTD + col];
    }
    red[tid] = acc; __syncthreads();
    if (tid < 128)
        atomicAdd(&ws[WS_CENTER + b * OUTD + col], red[tid] + red[tid + 128]);
}

// =====================================================================
// K5: out[b,0,col] = center[b][col] / Z[b, col>>5]
// =====================================================================
__global__ void gat_center_kernel(float* __restrict__ out,
                                  const float* __restrict__ ws) {
    const int b = blockIdx.x;       // 32 blocks
    const int col = threadIdx.x;    // 128 threads
    const float Z = ws[WS_STATS + (b * Hh + (col >> 5)) * 2 + 1];
    out[(size_t)b * Nsz * OUTD + col] = ws[WS_CENTER + b * OUTD + col] / Z;
}

// =====================================================================
extern "C" void kernel_launch(void* const* d_in, const int* in_sizes, int n_in,
                              void* d_out, int out_size, void* d_ws, size_t ws_size,
                              hipStream_t stream) {
    const float* x   = (const float*)d_in[0];
    const float* Wl  = (const float*)d_in[1];
    const float* bl  = (const float*)d_in[2];
    const float* Wr  = (const float*)d_in[3];
    const float* br  = (const float*)d_in[4];
    const float* att = (const float*)d_in[5];
    float* out = (float*)d_out;
    float* ws  = (float*)d_ws;

    (void)in_sizes; (void)n_in; (void)out_size; (void)ws_size;

    // K1: x_l0
    gat_xl0_kernel<<<Bsz, 128, 0, stream>>>(x, Wl, bl, ws);

    // K2: main WMMA GEMM + fused logits.  96 KB dynamic LDS.
    const int nblocks = (Bsz * Nsz) / 64;               // 4096
    const size_t smem = (64 * IND + IND * OUTD) * sizeof(float);  // 96 KB
    gat_gemm_kernel<<<nblocks, 256, smem, stream>>>(x, Wr, br, att, out, ws);

    // K3: softmax stats + zero center
    gat_stats_kernel<<<Bsz * Hh, 256, 0, stream>>>(ws);

    // K4: weighted aggregation
    gat_agg_kernel<<<Bsz * 16, 256, 0, stream>>>(out, ws);

    // K5: write center row
    gat_center_kernel<<<Bsz, 128, 0, stream>>>(out, ws);
}